// Postprocess_24575802867980
// MI455X (gfx1250) — compile-verified
//
#include <hip/hip_runtime.h>
#include <stdint.h>

// NMS postprocess for MI455X (gfx1250).
// Kernel 1: embarrassingly parallel gather/clip/key build (memory-bound, ~3MB).
// Kernel 2: ONE workgroup (32 wave32 waves, 1024 thr) holding the whole problem
//           in CDNA5's 320KB LDS: 64KB sort keys + 128KB sorted boxes + 1KB keep
//           bitmask. Bitonic sort in LDS, async global->LDS DMA gather of the
//           permuted boxes (global_load_async_to_lds_b128 + s_wait_asynccnt),
//           then greedy suppression that SKIPS dead boxes via a bitmask scan
//           (2 barriers per *kept* box instead of 8192 barrier iterations) and
//           clears suppressed bits with one wave32 ballot per 32-box word.

#define MBOX      8192
#define NCLS      80
#define NTHREADS  1024
#define EPT       (MBOX / NTHREADS)   // 8 elements per thread
#define NWORDS    (MBOX / 32)         // 256 keep-mask words
#define CONF_THR  0.25f
#define IOU_THR   0.45f

__global__ void prep_kernel(const float* __restrict__ boxes_raw,   // (M,4) [y1,x1,y2,x2]
                            const float* __restrict__ scores,      // (C,M)
                            const int*   __restrict__ indices,     // (M,3) [0,cls,idx]
                            const int*   __restrict__ out_h,
                            const int*   __restrict__ out_w,
                            uint64_t*    __restrict__ key_ws,
                            float4*      __restrict__ box_ws,
                            float*       __restrict__ conf_ws,
                            int*         __restrict__ cls_ws)
{
    int i = blockIdx.x * blockDim.x + threadIdx.x;
    if (i >= MBOX) return;

    int   cls  = indices[3 * i + 1];
    int   bidx = indices[3 * i + 2];
    float conf = scores[cls * MBOX + bidx];

    float y1 = boxes_raw[4 * bidx + 0];
    float x1 = boxes_raw[4 * bidx + 1];
    float y2 = boxes_raw[4 * bidx + 2];
    float x2 = boxes_raw[4 * bidx + 3];

    float W = (float)out_w[0], H = (float)out_h[0];
    float bx1 = fminf(fmaxf(x1, 0.f), W);
    float by1 = fminf(fmaxf(y1, 0.f), H);
    float bx2 = fminf(fmaxf(x2, 0.f), W);
    float by2 = fminf(fmaxf(y2, 0.f), H);

    bool valid = (conf > CONF_THR) &&
                 (cls == 2 || cls == 3 || cls == 5 || cls == 7);
    float s = valid ? conf : -1.0f;

    // Monotonic float->u32 key, descending order, stable tie-break by index
    // (matches jnp.argsort(-s) stable semantics).
    unsigned b    = __float_as_uint(s);
    unsigned asc  = (b & 0x80000000u) ? ~b : (b | 0x80000000u);
    unsigned desc = ~asc;
    uint64_t key  = ((uint64_t)desc << 32) | (valid ? 0x2000u : 0u) | (unsigned)i;

    key_ws[i]  = key;
    box_ws[i]  = make_float4(bx1, by1, bx2, by2);  // [x1,y1,x2,y2] clipped
    conf_ws[i] = conf;
    cls_ws[i]  = cls;
}

// Uniform (all-threads-identical) scan for the next surviving sorted index
// >= start. Pure LDS broadcast reads; returns MBOX when none remain.
__device__ __forceinline__ unsigned find_next(const unsigned* keepm, unsigned start)
{
    if (start >= MBOX) return MBOX;
    unsigned w = start >> 5;
    unsigned m = keepm[w] & (0xFFFFFFFFu << (start & 31));
    while (m == 0u) {
        if (++w >= NWORDS) return MBOX;
        m = keepm[w];
    }
    return (w << 5) + (unsigned)__ffs(m) - 1u;
}

__global__ __launch_bounds__(NTHREADS)
void sortnms_kernel(const uint64_t* __restrict__ key_ws,
                    const float4*   __restrict__ box_ws,
                    const float*    __restrict__ conf_ws,
                    const int*      __restrict__ cls_ws,
                    float*          __restrict__ out)
{
    extern __shared__ unsigned char smem[];
    uint64_t* sk    = (uint64_t*)smem;                                       // 64 KB
    float4*   sbox  = (float4*)(smem + sizeof(uint64_t) * MBOX);             // 128 KB
    unsigned* keepm = (unsigned*)(smem + sizeof(uint64_t) * MBOX
                                       + sizeof(float4) * MBOX);             // 1 KB

    const unsigned t = threadIdx.x;

    // ---- load keys into LDS ----
    for (unsigned e = 0; e < EPT; ++e)
        sk[t + e * NTHREADS] = key_ws[t + e * NTHREADS];

    // ---- bitonic sort of 8192 u64 keys in LDS (ascending = score descending) ----
    for (unsigned k = 2; k <= MBOX; k <<= 1) {
        for (unsigned j = k >> 1; j > 0; j >>= 1) {
            __syncthreads();
            for (unsigned e = 0; e < EPT; ++e) {
                unsigned i   = t + e * NTHREADS;
                unsigned ixj = i ^ j;
                if (ixj > i) {
                    uint64_t a = sk[i], c = sk[ixj];
                    bool up = ((i & k) == 0);
                    if ((a > c) == up) { sk[i] = c; sk[ixj] = a; }
                }
            }
        }
    }
    __syncthreads();

    // ---- zero keep mask, then async DMA-gather sorted boxes into LDS ----
    if (t < NWORDS) keepm[t] = 0u;

    for (unsigned e = 0; e < EPT; ++e) {
        unsigned s    = t + e * NTHREADS;
        unsigned orig = (unsigned)sk[s] & 0x1FFFu;
        unsigned lds_addr = (unsigned)(uintptr_t)(&sbox[s]);          // LDS byte addr
        unsigned long long gaddr =
            (unsigned long long)(uintptr_t)(&box_ws[orig]);           // global addr
        // CDNA5 async global->LDS DMA (ISA 15.18.3 op 98), tracked by ASYNCcnt.
        asm volatile("global_load_async_to_lds_b128 %0, %1, off"
                     :: "v"(lds_addr), "v"(gaddr) : "memory");
    }

    __syncthreads();   // keepm zeros visible to everyone

    // set keep bit for valid sorted entries while the DMA is in flight
    for (unsigned e = 0; e < EPT; ++e) {
        unsigned s = t + e * NTHREADS;
        if ((sk[s] >> 13) & 1ull)
            atomicOr(&keepm[s >> 5], 1u << (s & 31));
    }

#if __has_builtin(__builtin_amdgcn_s_wait_asynccnt)
    __builtin_amdgcn_s_wait_asynccnt(0);
#else
    asm volatile("s_wait_asynccnt 0" ::: "memory");
#endif
    __syncthreads();   // boxes + keep bits fully resident in LDS

    // ---- greedy NMS: jump from kept box to kept box via bitmask scan ----
    const unsigned wave = t >> 5, lane = t & 31;
    unsigned i = find_next(keepm, 0);
    while (i < MBOX - 1) {
        float4 bi = sbox[i];
        float  ai = fmaxf(bi.z - bi.x, 0.f) * fmaxf(bi.w - bi.y, 0.f);

#if __has_builtin(__builtin_amdgcn_ballot_w32)
        // wave32 ballot path: each wave owns disjoint 32-box mask words;
        // one plain LDS store per word instead of per-bit atomics.
        for (unsigned word = (i >> 5) + wave; word < NWORDS; word += 32) {
            unsigned cur = keepm[word];
            unsigned j   = (word << 5) + lane;
            bool sup = false;
            if (j > i && ((cur >> lane) & 1u)) {
                float4 bj = sbox[j];
                float ix1 = fmaxf(bi.x, bj.x), iy1 = fmaxf(bi.y, bj.y);
                float ix2 = fminf(bi.z, bj.z), iy2 = fminf(bi.w, bj.w);
                float inter = fmaxf(ix2 - ix1, 0.f) * fmaxf(iy2 - iy1, 0.f);
                float aj = fmaxf(bj.z - bj.x, 0.f) * fmaxf(bj.w - bj.y, 0.f);
                float iou = inter / (ai + aj - inter + 1e-9f);
                sup = (iou > IOU_THR);
            }
            unsigned mask = __builtin_amdgcn_ballot_w32(sup);
            if (lane == 0 && mask) keepm[word] = cur & ~mask;
        }
#else
        for (unsigned j = i + 1 + t; j < MBOX; j += NTHREADS) {
            if (!((keepm[j >> 5] >> (j & 31)) & 1u)) continue;
            float4 bj = sbox[j];
            float ix1 = fmaxf(bi.x, bj.x), iy1 = fmaxf(bi.y, bj.y);
            float ix2 = fminf(bi.z, bj.z), iy2 = fminf(bi.w, bj.w);
            float inter = fmaxf(ix2 - ix1, 0.f) * fmaxf(iy2 - iy1, 0.f);
            float aj = fmaxf(bj.z - bj.x, 0.f) * fmaxf(bj.w - bj.y, 0.f);
            float iou = inter / (ai + aj - inter + 1e-9f);
            if (iou > IOU_THR)
                atomicAnd(&keepm[j >> 5], ~(1u << (j & 31)));
        }
#endif
        __syncthreads();                      // suppressions visible
        unsigned inext = find_next(keepm, i + 1);
        __syncthreads();                      // all done scanning before next clears
        i = inext;
    }
    __syncthreads();

    // ---- scatter outputs back to original order ----
    // d_out layout (floats): [boxes 4M][conf M][class M][keep M]
    for (unsigned e = 0; e < EPT; ++e) {
        unsigned s    = t + e * NTHREADS;
        unsigned orig = (unsigned)sk[s] & 0x1FFFu;
        float    m    = ((keepm[s >> 5] >> (s & 31)) & 1u) ? 1.0f : 0.0f;
        float4   bb   = sbox[s];
        out[orig * 4 + 0] = bb.x * m;
        out[orig * 4 + 1] = bb.y * m;
        out[orig * 4 + 2] = bb.z * m;
        out[orig * 4 + 3] = bb.w * m;
        out[4 * MBOX + orig] = conf_ws[orig] * m;
        out[5 * MBOX + orig] = (float)cls_ws[orig];
        out[6 * MBOX + orig] = m;
    }
}

extern "C" void kernel_launch(void* const* d_in, const int* in_sizes, int n_in,
                              void* d_out, int out_size, void* d_ws, size_t ws_size,
                              hipStream_t stream)
{
    const float* boxes_raw = (const float*)d_in[0];   // (1,M,4) f32
    const float* scores    = (const float*)d_in[1];   // (1,C,M) f32
    const int*   indices   = (const int*)d_in[2];     // (1,M,3) i32
    const int*   out_h     = (const int*)d_in[3];     // scalar
    const int*   out_w     = (const int*)d_in[4];     // scalar
    float* out = (float*)d_out;

    // workspace carve: 256 KB total
    unsigned char* ws = (unsigned char*)d_ws;
    uint64_t* key_ws  = (uint64_t*)(ws);                          // 64 KB
    float4*   box_ws  = (float4*)(ws + 65536);                    // 128 KB
    float*    conf_ws = (float*)(ws + 65536 + 131072);            // 32 KB
    int*      cls_ws  = (int*)(ws + 65536 + 131072 + 32768);      // 32 KB

    prep_kernel<<<MBOX / 256, 256, 0, stream>>>(
        boxes_raw, scores, indices, out_h, out_w,
        key_ws, box_ws, conf_ws, cls_ws);

    // 64KB keys + 128KB boxes + 1KB mask = 193 KB dynamic LDS
    // (fits CDNA5's 320KB/WGP; far beyond CDNA4's 64KB)
    size_t smem = sizeof(uint64_t) * MBOX + sizeof(float4) * MBOX
                + NWORDS * sizeof(unsigned);
    sortnms_kernel<<<1, NTHREADS, smem, stream>>>(
        key_ws, box_ws, conf_ws, cls_ws, out);
}